// Model_14482629722734
// MI455X (gfx1250) — compile-verified
//
#include <hip/hip_runtime.h>
#include <math.h>

// Problem constants
#define BB   32
#define LL   1024
#define NN   64
#define DPE  16
#define DK   96
#define NHID 128
#define PLEN 96

typedef float v2f __attribute__((ext_vector_type(2)));
typedef float v8f __attribute__((ext_vector_type(8)));

#if defined(__HIP_DEVICE_COMPILE__) && __has_builtin(__builtin_amdgcn_wmma_f32_16x16x4_f32)
#define USE_WMMA 1
#else
#define USE_WMMA 0
#endif

// -------------------------------------------------------------------------
// Kernel P: precompute sinusoidal positional encodings into workspace.
// ws[0 .. 16384)        : pe_t [1024][16]   (t = l / L)
// ws[16384 .. 17408)    : pe_v [64][16]     (t = n)
// pe[k] = sin(t / ts[k])        for k < 8
//       = cos(t / ts[k-8])      for k >= 8,   ts[j] = 1024^(j/7) = 2^(10j/7)
// -------------------------------------------------------------------------
__global__ __launch_bounds__(256) void pe_kernel(float* __restrict__ ws) {
    int idx = blockIdx.x * 256 + threadIdx.x;
    const int total = LL * DPE + NN * DPE;
    if (idx >= total) return;
    float t;
    int k;
    if (idx < LL * DPE) {
        int l = idx >> 4; k = idx & 15;
        t = (float)l * (1.0f / (float)LL);
    } else {
        int i2 = idx - LL * DPE;
        int n = i2 >> 4; k = i2 & 15;
        t = (float)n;
    }
    int j = k & 7;
    float ts = exp2f((10.0f * (float)j) / 7.0f);
    float s = t / ts;
    ws[idx] = (k < 8) ? sinf(s) : cosf(s);
}

// -------------------------------------------------------------------------
// Kernel A: masked reduction over L + feature assembly.
// One block per batch b. 256 threads = 64 vars (n) x 4 L-slices.
// Produces feat [2048][96] in workspace.
// -------------------------------------------------------------------------
__global__ __launch_bounds__(256) void reduce_feat_kernel(
    const float* __restrict__ x,      // [B,L,N]
    const float* __restrict__ w_val,  // [16]
    const float* __restrict__ b_val,  // [16]
    const float* __restrict__ pet,    // [1024][16]
    const float* __restrict__ pev,    // [64][16]
    float* __restrict__ feat)         // [2048][96]
{
    __shared__ float s_mt[4 * 64 * 16];   // per-slice partial pe_t sums
    __shared__ float s_cnt[4 * 64];
    __shared__ float s_sx[4 * 64];
    __shared__ float s_red[64 * 34];      // per-n: cnt, sx, mt[16], cnt*pe_v[16]
    __shared__ float s_sum[34];           // sums over n

    const int tid = threadIdx.x;
    const int b   = blockIdx.x;
    const int n   = tid & 63;
    const int sl  = tid >> 6;

    float cnt = 0.0f, sx = 0.0f;
    float mt[16];
    #pragma unroll
    for (int k = 0; k < 16; ++k) mt[k] = 0.0f;

    const float* xb = x + (size_t)b * LL * NN;
    const int lbase = sl * (LL / 4);
    for (int i = 0; i < LL / 4; ++i) {
        int l = lbase + i;
        float v = xb[(size_t)l * NN + n];
        float m = (v != 0.0f) ? 1.0f : 0.0f;
        cnt += m;
        sx  += v;                 // mask*x == x since mask = (x != 0)
        const float* pr = pet + (size_t)l * 16;
        #pragma unroll
        for (int k = 0; k < 16; ++k) mt[k] = fmaf(m, pr[k], mt[k]);
    }

    {
        int base = (sl * 64 + n) * 16;
        #pragma unroll
        for (int k = 0; k < 16; ++k) s_mt[base + k] = mt[k];
        s_cnt[sl * 64 + n] = cnt;
        s_sx[sl * 64 + n]  = sx;
    }
    __syncthreads();

    float cnt_n = 0.0f, sx_n = 0.0f;
    float pv[16];
    if (tid < 64) {
        cnt_n = s_cnt[n] + s_cnt[64 + n] + s_cnt[128 + n] + s_cnt[192 + n];
        sx_n  = s_sx[n]  + s_sx[64 + n]  + s_sx[128 + n]  + s_sx[192 + n];
        #pragma unroll
        for (int k = 0; k < 16; ++k) {
            mt[k] = s_mt[(0 * 64 + n) * 16 + k] + s_mt[(1 * 64 + n) * 16 + k]
                  + s_mt[(2 * 64 + n) * 16 + k] + s_mt[(3 * 64 + n) * 16 + k];
        }
        #pragma unroll
        for (int k = 0; k < 16; ++k) pv[k] = pev[n * 16 + k];
        s_red[n * 34 + 0] = cnt_n;
        s_red[n * 34 + 1] = sx_n;
        #pragma unroll
        for (int k = 0; k < 16; ++k) s_red[n * 34 + 2 + k]  = mt[k];
        #pragma unroll
        for (int k = 0; k < 16; ++k) s_red[n * 34 + 18 + k] = cnt_n * pv[k];
    }
    __syncthreads();

    if (tid < 34) {
        float s = 0.0f;
        for (int q = 0; q < 64; ++q) s += s_red[q * 34 + tid];
        s_sum[tid] = s;
    }
    __syncthreads();

    if (tid < 64) {
        const float cnt_tot = s_sum[0];
        const float ssx     = s_sum[1];
        const float inv_tot = 1.0f / fmaxf(cnt_tot, 1.0f);
        const float has     = (cnt_n > 0.0f) ? 1.0f : 0.0f;
        const float inv_n   = 1.0f / fmaxf(cnt_n, 1.0f);
        float* fr = feat + (size_t)(b * 64 + n) * 96;
        #pragma unroll
        for (int k = 0; k < 16; ++k) {
            fr[k]      = has * (s_sum[2 + k] * inv_tot);          // f_rep, pe_t part
            fr[48 + k] = has * (mt[k] * inv_n);                   // unit_mean, pe_t part
        }
        #pragma unroll
        for (int k = 0; k < 16; ++k) {
            float fp = (ssx * w_val[k] + cnt_tot * b_val[k]) * inv_tot;
            fr[16 + k]      = has * fp;                           // f_rep, value part
            fr[48 + 16 + k] = has * ((sx_n * w_val[k] + cnt_n * b_val[k]) * inv_n);
        }
        #pragma unroll
        for (int k = 0; k < 16; ++k) {
            fr[32 + k]      = has * (s_sum[18 + k] * inv_tot);    // f_rep, pe_v part
            fr[48 + 32 + k] = has * (cnt_n * pv[k] * inv_n);      // unit_mean = pe_v (cnt cancels)
        }
    }
}

// -------------------------------------------------------------------------
// Kernel B: fused MLP chain on 16-row tiles using exact f32 WMMA (16x16x4).
// 8 waves per block; wave w owns output cols [16w, 16w+16).
// Fragment layouts per CDNA5 ISA 7.12.2 (32-bit A 16x4, B 4x16, C/D 16x16):
//   lane = 16*kh + n  (kh = lane>>4, n = lane&15)
//   A: a.x = A[m=n][k0+2kh], a.y = A[n][k0+2kh+1]
//   B: b.x = B[k0+2kh][col], b.y = B[k0+2kh+1][col]
//   C/D: c[r] = C[r + 8*kh][col]
// -------------------------------------------------------------------------
__global__ __launch_bounds__(256) void mlp_kernel(
    const float* __restrict__ feat,   // [2048][96]
    const float* __restrict__ w_map,  // [96][128]
    const float* __restrict__ b_map,  // [128]
    const float* __restrict__ w1,     // [128][128]
    const float* __restrict__ b1,     // [128]
    const float* __restrict__ w2,     // [128][96]
    const float* __restrict__ b2,     // [96]
    float* __restrict__ out)          // [32][96][64]
{
    __shared__ float fa[16 * 97];     // feat tile, padded stride
    __shared__ float h1s[16 * 129];   // lin_map output tile
    __shared__ float h2s[16 * 129];   // relu(mlp1) tile

    const int tid = threadIdx.x;
    const int m0  = blockIdx.x * 16;

    for (int idx = tid; idx < 16 * 96; idx += 256) {
        int r = idx / 96, c = idx - r * 96;
        fa[r * 97 + c] = feat[(size_t)(m0 + r) * 96 + c];
    }
    __syncthreads();

#if USE_WMMA
    const int lane = tid & 31;
    const int w    = tid >> 5;
    const int nn   = lane & 15;
    const int kh   = lane >> 4;
    const int col  = w * 16 + nn;

    v2f a, bv;
    v8f acc;

    // ---- Stage 1: h1 = feat @ w_map + b_map  (K = 96) ----
    {
        float bb = b_map[col];
        #pragma unroll
        for (int r = 0; r < 8; ++r) acc[r] = bb;
    }
    #pragma unroll
    for (int k0 = 0; k0 < 96; k0 += 4) {
        a.x  = fa[nn * 97 + k0 + 2 * kh];
        a.y  = fa[nn * 97 + k0 + 2 * kh + 1];
        bv.x = w_map[(k0 + 2 * kh) * 128 + col];
        bv.y = w_map[(k0 + 2 * kh + 1) * 128 + col];
        acc = __builtin_amdgcn_wmma_f32_16x16x4_f32(false, a, false, bv,
                                                    (short)0, acc, false, false);
    }
    #pragma unroll
    for (int r = 0; r < 8; ++r) h1s[(r + 8 * kh) * 129 + col] = acc[r];
    __syncthreads();

    // ---- Stage 2: h2 = relu(h1 @ w1 + b1)  (K = 128) ----
    {
        float bb = b1[col];
        #pragma unroll
        for (int r = 0; r < 8; ++r) acc[r] = bb;
    }
    #pragma unroll
    for (int k0 = 0; k0 < 128; k0 += 4) {
        a.x  = h1s[nn * 129 + k0 + 2 * kh];
        a.y  = h1s[nn * 129 + k0 + 2 * kh + 1];
        bv.x = w1[(k0 + 2 * kh) * 128 + col];
        bv.y = w1[(k0 + 2 * kh + 1) * 128 + col];
        acc = __builtin_amdgcn_wmma_f32_16x16x4_f32(false, a, false, bv,
                                                    (short)0, acc, false, false);
    }
    #pragma unroll
    for (int r = 0; r < 8; ++r)
        h2s[(r + 8 * kh) * 129 + col] = fmaxf(acc[r], 0.0f);
    __syncthreads();

    // ---- Stage 3: out = h2 @ w2 + b2  (K = 128, 96 cols -> waves 0..5) ----
    if (w < 6) {
        const int colp = w * 16 + nn;
        {
            float bb = b2[colp];
            #pragma unroll
            for (int r = 0; r < 8; ++r) acc[r] = bb;
        }
        #pragma unroll
        for (int k0 = 0; k0 < 128; k0 += 4) {
            a.x  = h2s[nn * 129 + k0 + 2 * kh];
            a.y  = h2s[nn * 129 + k0 + 2 * kh + 1];
            bv.x = w2[(k0 + 2 * kh) * 96 + colp];
            bv.y = w2[(k0 + 2 * kh + 1) * 96 + colp];
            acc = __builtin_amdgcn_wmma_f32_16x16x4_f32(false, a, false, bv,
                                                        (short)0, acc, false, false);
        }
        #pragma unroll
        for (int r = 0; r < 8; ++r) {
            int M    = m0 + r + 8 * kh;
            int bidx = M >> 6;
            int nv   = M & 63;
            out[(size_t)bidx * (PLEN * NN) + (size_t)colp * NN + nv] = acc[r];
        }
    }
#else
    // Scalar fallback (kept only so the source compiles if the f32 WMMA
    // builtin is absent; if the histogram shows wmma==0 we switch shapes).
    for (int idx = tid; idx < 16 * 128; idx += 256) {
        int r = idx >> 7, j = idx & 127;
        float s = b_map[j];
        for (int k = 0; k < 96; ++k) s = fmaf(fa[r * 97 + k], w_map[k * 128 + j], s);
        h1s[r * 129 + j] = s;
    }
    __syncthreads();
    for (int idx = tid; idx < 16 * 128; idx += 256) {
        int r = idx >> 7, j = idx & 127;
        float s = b1[j];
        for (int k = 0; k < 128; ++k) s = fmaf(h1s[r * 129 + k], w1[k * 128 + j], s);
        h2s[r * 129 + j] = fmaxf(s, 0.0f);
    }
    __syncthreads();
    for (int idx = tid; idx < 16 * 96; idx += 256) {
        int r = idx / 96, p = idx - r * 96;
        float s = b2[p];
        for (int k = 0; k < 128; ++k) s = fmaf(h2s[r * 129 + k], w2[k * 96 + p], s);
        int M = m0 + r;
        out[(size_t)(M >> 6) * (PLEN * NN) + (size_t)p * NN + (M & 63)] = s;
    }
#endif
}

// -------------------------------------------------------------------------
// Launch
// Workspace layout (floats): [0,16384) pe_t | [16384,17408) pe_v |
//                            [17408, 17408+196608) feat
// -------------------------------------------------------------------------
extern "C" void kernel_launch(void* const* d_in, const int* in_sizes, int n_in,
                              void* d_out, int out_size, void* d_ws, size_t ws_size,
                              hipStream_t stream) {
    const float* x     = (const float*)d_in[0];
    const float* w_val = (const float*)d_in[1];
    const float* b_val = (const float*)d_in[2];
    const float* w_map = (const float*)d_in[3];
    const float* b_map = (const float*)d_in[4];
    const float* w1    = (const float*)d_in[5];
    const float* b1    = (const float*)d_in[6];
    const float* w2    = (const float*)d_in[7];
    const float* b2    = (const float*)d_in[8];

    float* ws   = (float*)d_ws;
    float* pet  = ws;              // 1024*16
    float* pev  = ws + 16384;      // 64*16
    float* feat = ws + 17408;      // 2048*96
    float* out  = (float*)d_out;

    pe_kernel<<<68, 256, 0, stream>>>(ws);
    reduce_feat_kernel<<<BB, 256, 0, stream>>>(x, w_val, b_val, pet, pev, feat);
    mlp_kernel<<<(BB * NN) / 16, 256, 0, stream>>>(feat, w_map, b_map, w1, b1, w2, b2, out);
}